// Quantizing_1580547970990
// MI455X (gfx1250) — compile-verified
//
#include <hip/hip_runtime.h>
#include <stdint.h>

// ---------------------------------------------------------------------------
// Fused VQ codebook search for MI455X (gfx1250, wave32, WMMA + async-to-LDS).
//   dist = ||h||^2 + ||w||^2 - 2 h.w ; argmin over 4096 codewords; gather.
// GEMM core: M=65536, N=4096, K=64 via V_WMMA_F32_16X16X4_F32 (exact fp32 so
// argmin ties match the fp32 reference). ||h||^2 is row-constant -> dropped.
// Codebook tiles are double-buffered in LDS via GLOBAL_LOAD_ASYNC_TO_LDS_B128
// (ASYNCcnt), shared by all 8 waves of the block: cuts L2 reads of W from
// 2 GB -> 256 MB and feeds B fragments from 64-bank LDS (ds_load_b64).
// ---------------------------------------------------------------------------

typedef float v2f __attribute__((ext_vector_type(2)));
typedef float v8f __attribute__((ext_vector_type(8)));

#define N_ROWS 65536
#define NQ     4096
#define DIM    64
#define KSTEPS 16          // 64 / 4 (K per f32 WMMA)
#define TILES_PER_WAVE 2   // two 16-row C tiles per wave -> 32 rows/wave
#define QT     64          // codewords per LDS tile (16 KB)
#define TILE_FLOATS (QT * DIM)
#define TILE_BYTES  (TILE_FLOATS * 4)

// --- Pre-pass: w_sq[q] = sum_k weight[q][k]^2  (4096 floats into d_ws) -----
__global__ void vq_wsq_kernel(const float* __restrict__ w,
                              float* __restrict__ wsq) {
    int q = blockIdx.x * blockDim.x + threadIdx.x;
    const float* row = w + (size_t)q * DIM;
    float s = 0.f;
#pragma unroll
    for (int k = 0; k < DIM; ++k) s += row[k] * row[k];
    wsq[q] = s;
}

// Async-stage one 64x64 f32 codebook tile into LDS.
// 256 threads x 4 x b128 = 16 KB; tracked on ASYNCcnt.
__device__ __forceinline__ void stage_tile(const float* __restrict__ wsrc,
                                           uint32_t lds_off, int tid) {
#pragma unroll
    for (int k = 0; k < 4; ++k) {
        uint32_t off = (uint32_t)(tid * 16 + k * 4096);
        uint64_t ga  = (uint64_t)(uintptr_t)wsrc + off;
        uint32_t la  = lds_off + off;
        asm volatile("global_load_async_to_lds_b128 %0, %1, off"
                     :: "v"(la), "v"(ga) : "memory");
    }
}

// --- Main fused GEMM + argmin + gather -------------------------------------
__global__ void __launch_bounds__(256)
vq_search_kernel(const float* __restrict__ x,
                 const float* __restrict__ w,
                 const float* __restrict__ wsq,
                 float* __restrict__ qdata,   // (N_ROWS, DIM)
                 float* __restrict__ qidx) {  // (N_ROWS,) written as float
    __shared__ __align__(16) float smem[2 * TILE_FLOATS];   // 32 KB, 2 buffers

    const int tid  = threadIdx.x;
    const int lane = tid & 31;
    const int wave = (int)((blockIdx.x * blockDim.x + tid) >> 5);
    const int rowbase = wave * (16 * TILES_PER_WAVE);
    const int col  = lane & 15;          // N (and M for A fragments)
    const int koff = (lane >> 4) << 1;   // 0 or 2: K sub-offset per half-wave

    // LDS byte offset of smem: generic LDS pointers are {aperture_hi, offset}.
    const uint32_t lds_base = (uint32_t)(uintptr_t)(const void*)smem;

    // A fragments: a[t][s] = x[rowbase+16t+col][4s+koff .. +1], resident all
    // kernel long (64 VGPRs) so the q loop only streams B through LDS.
    v2f a[TILES_PER_WAVE][KSTEPS];
#pragma unroll
    for (int t = 0; t < TILES_PER_WAVE; ++t) {
        const float* xr = x + (size_t)(rowbase + t * 16 + col) * DIM + koff;
#pragma unroll
        for (int s = 0; s < KSTEPS; ++s)
            a[t][s] = *(const v2f*)(xr + 4 * s);
    }

    float minv[TILES_PER_WAVE][8];
    int   mini[TILES_PER_WAVE][8];
#pragma unroll
    for (int t = 0; t < TILES_PER_WAVE; ++t)
#pragma unroll
        for (int r = 0; r < 8; ++r) { minv[t][r] = 3.4e38f; mini[t][r] = 0; }

    // Kick off tile 0 into buffer 0.
    stage_tile(w, lds_base, tid);

    for (int it = 0; it < NQ / QT; ++it) {
        // Publish tile `it`: my async copies done, then block-wide barrier.
        asm volatile("s_wait_asynccnt 0x0" ::: "memory");
        __syncthreads();

        // Prefetch tile it+1 into the other buffer while we compute.
        if (it + 1 < NQ / QT)
            stage_tile(w + (size_t)(it + 1) * TILE_FLOATS,
                       lds_base + (uint32_t)(((it + 1) & 1) * TILE_BYTES), tid);

        const float* wt = smem + (it & 1) * TILE_FLOATS;
        const int qbase = it * QT;

#pragma unroll
        for (int qs = 0; qs < QT; qs += 16) {
            // B fragments (W^T 16-col tile) from LDS: ds_load_b64 per step.
            v2f b[KSTEPS];
            const float* wr = wt + (qs + col) * DIM + koff;
#pragma unroll
            for (int s = 0; s < KSTEPS; ++s)
                b[s] = *(const v2f*)(wr + 4 * s);
            const float wsq_c = wsq[qbase + qs + col];

#pragma unroll
            for (int t = 0; t < TILES_PER_WAVE; ++t) {
                v8f acc = {};
#pragma unroll
                for (int s = 0; s < KSTEPS; ++s)
                    acc = __builtin_amdgcn_wmma_f32_16x16x4_f32(
                        /*neg_a=*/false, a[t][s], /*neg_b=*/false, b[s],
                        /*c_mod=*/(short)0, acc,
                        /*reuse_a=*/false, /*reuse_b=*/false);
                // score = ||w||^2 - 2 h.w  (||h||^2 row-constant, dropped)
#pragma unroll
                for (int r = 0; r < 8; ++r) {
                    float sc = wsq_c - 2.0f * acc[r];
                    if (sc < minv[t][r]) {
                        minv[t][r] = sc;
                        mini[t][r] = qbase + qs + col;
                    }
                }
            }
        }
        // All waves done reading this buffer before it is overwritten.
        __syncthreads();
    }

    // Cross-column argmin: butterfly over 16 lanes; masks 8..1 keep the two
    // 16-lane halves (rows r and r+8) independent on wave32. Ties -> lower q.
#pragma unroll
    for (int t = 0; t < TILES_PER_WAVE; ++t) {
#pragma unroll
        for (int r = 0; r < 8; ++r) {
            float v = minv[t][r];
            int   i = mini[t][r];
#pragma unroll
            for (int m = 8; m >= 1; m >>= 1) {
                float ov = __shfl_xor(v, m, 32);
                int   oi = __shfl_xor(i, m, 32);
                if (ov < v || (ov == v && oi < i)) { v = ov; i = oi; }
            }
            const int m0 = rowbase + t * 16 + r;   // result in lanes 0..15
            const int m1 = m0 + 8;                 // result in lanes 16..31
            const int iA = __shfl(i, 0, 32);
            const int iB = __shfl(i, 16, 32);
            // Cooperative gather: 32 lanes x float2 = 64 floats per row.
            v2f ga = *(const v2f*)(w + (size_t)iA * DIM + 2 * lane);
            *(v2f*)(qdata + (size_t)m0 * DIM + 2 * lane) = ga;
            v2f gb = *(const v2f*)(w + (size_t)iB * DIM + 2 * lane);
            *(v2f*)(qdata + (size_t)m1 * DIM + 2 * lane) = gb;
            if (lane == 0) {
                qidx[m0] = (float)iA;
                qidx[m1] = (float)iB;
            }
        }
    }
}

extern "C" void kernel_launch(void* const* d_in, const int* in_sizes, int n_in,
                              void* d_out, int out_size, void* d_ws, size_t ws_size,
                              hipStream_t stream) {
    const float* x = (const float*)d_in[0];   // (65536, 64) f32
    const float* w = (const float*)d_in[1];   // (4096, 64) f32
    float* out  = (float*)d_out;              // qdata (65536*64) then qidx (65536)
    float* wsq  = (float*)d_ws;               // 4096 floats scratch

    vq_wsq_kernel<<<NQ / 256, 256, 0, stream>>>(w, wsq);

    // 65536 rows / 32 rows-per-wave = 2048 waves = 65536 threads, 256/block.
    const int threads = (N_ROWS / (16 * TILES_PER_WAVE)) * 32;
    vq_search_kernel<<<threads / 256, 256, 0, stream>>>(
        x, w, wsq, out, out + (size_t)N_ROWS * DIM);
}